// PoolSAGE_644245095092
// MI455X (gfx1250) — compile-verified
//
#include <hip/hip_runtime.h>

typedef __attribute__((ext_vector_type(2))) float v2f;
typedef __attribute__((ext_vector_type(8))) float v8f;

#define LDSTR 132   // 128 + 4 pad floats: 528B row stride, 8B aligned, bank-conflict free

// ---------------- utility kernels ----------------

__global__ void fill_f32(float* __restrict__ p, long long n, float v) {
    long long i = (long long)blockIdx.x * blockDim.x + threadIdx.x;
    if (i < n) p[i] = v;
}

__global__ void fill_i32(int* __restrict__ p, int n, int v) {
    int i = blockIdx.x * blockDim.x + threadIdx.x;
    if (i < n) p[i] = v;
}

__global__ void copy_f4(const float4* __restrict__ s, float4* __restrict__ d, int n4) {
    int i = blockIdx.x * blockDim.x + threadIdx.x;
    if (i < n4) d[i] = s[i];
}

// W: [128][dout] row-major  ->  WT: [dout][128]
__global__ void transpose_w(const float* __restrict__ W, float* __restrict__ WT, int dout) {
    int i = blockIdx.x * blockDim.x + threadIdx.x;
    if (i >= 128 * dout) return;
    int k = i / dout;
    int c = i % dout;
    WT[c * 128 + k] = W[i];
}

// ---------------- CSR construction (once per launch; edges fixed across layers) ----

__global__ void hist_dst(const int* __restrict__ dst, int* __restrict__ counts, int E) {
    int e = blockIdx.x * blockDim.x + threadIdx.x;
    if (e < E) atomicAdd(&counts[dst[e]], 1);
}

// single-block exclusive scan: rowptr[0]=0, rowptr[i+1]=sum(counts[0..i])
__global__ __launch_bounds__(1024)
void scan_counts(const int* __restrict__ counts, int* __restrict__ rowptr, int n) {
    __shared__ int buf[1024];
    __shared__ int carry;
    const int tid = threadIdx.x;
    if (tid == 0) { carry = 0; rowptr[0] = 0; }
    __syncthreads();
    for (int base = 0; base < n; base += 1024) {
        int i = base + tid;
        int v = (i < n) ? counts[i] : 0;
        buf[tid] = v;
        __syncthreads();
        for (int off = 1; off < 1024; off <<= 1) {
            int t = (tid >= off) ? buf[tid - off] : 0;
            __syncthreads();
            buf[tid] += t;
            __syncthreads();
        }
        int incl = buf[tid] + carry;
        if (i < n) rowptr[i + 1] = incl;
        __syncthreads();
        if (tid == 1023) carry = incl;   // new carry = old carry + chunk total
        __syncthreads();
    }
}

__global__ void build_csr(const int* __restrict__ src, const int* __restrict__ dst,
                          const int* __restrict__ rowptr, int* __restrict__ cursor,
                          int* __restrict__ srcSorted, int E) {
    int e = blockIdx.x * blockDim.x + threadIdx.x;
    if (e >= E) return;
    int d   = dst[e];
    int pos = rowptr[d] + atomicAdd(&cursor[d], 1);
    srcSorted[pos] = src[e];
}

// ---------------- fused SAGE layer: out = x@Ws + mean_neigh(x)@Wn + b --------------
// grid.x = N/16 row tiles, 128 threads (4 waves).
// Phase 1: stage self tile + gather-aggregate neighbor tile into LDS (no atomics).
// Phase 2: WMMA f32 16x16x4 with transposed weights (contiguous b64 B-fragments).
template <int DOUT>
__global__ __launch_bounds__(128)
void sage_fused(const float* __restrict__ x, const int* __restrict__ rowptr,
                const int* __restrict__ srcs,
                const float* __restrict__ WsT, const float* __restrict__ WnT, // [DOUT][128]
                const float* __restrict__ bias, float* __restrict__ out) {
    __shared__ float As[16 * LDSTR];
    __shared__ float An[16 * LDSTR];

    const int tid     = threadIdx.x;
    const int wave    = tid >> 5;
    const int lane    = tid & 31;
    const int rowBase = blockIdx.x * 16;

    const float4* __restrict__ x4 = (const float4*)x;   // 32-bit element offsets only

    // stage self rows: 16 x 128 floats, 128 threads x 4 float4
#pragma unroll
    for (int i = 0; i < 4; ++i) {
        int id  = tid + i * 128;
        int row = id >> 5;
        int c4  = id & 31;
        float4 v = x4[(unsigned)(rowBase + row) * 32u + (unsigned)c4];
        float* p = &As[row * LDSTR + c4 * 4];
        p[0] = v.x; p[1] = v.y; p[2] = v.z; p[3] = v.w;
    }

    // gather-aggregate neighbors: each wave owns 4 rows (strided), float4 per lane,
    // 2x unrolled with dual accumulators for MLP
    for (int r = wave; r < 16; r += 4) {
        const int v  = rowBase + r;
        const int j0 = rowptr[v];
        const int j1 = rowptr[v + 1];
        float4 a0 = make_float4(0.f, 0.f, 0.f, 0.f);
        float4 a1 = make_float4(0.f, 0.f, 0.f, 0.f);
        int j = j0;
        for (; j + 1 < j1; j += 2) {
            int s0 = srcs[j];
            int s1 = srcs[j + 1];
            float4 t0 = x4[(unsigned)s0 * 32u + (unsigned)lane];
            float4 t1 = x4[(unsigned)s1 * 32u + (unsigned)lane];
            a0.x += t0.x; a0.y += t0.y; a0.z += t0.z; a0.w += t0.w;
            a1.x += t1.x; a1.y += t1.y; a1.z += t1.z; a1.w += t1.w;
        }
        if (j < j1) {
            int s0 = srcs[j];
            float4 t0 = x4[(unsigned)s0 * 32u + (unsigned)lane];
            a0.x += t0.x; a0.y += t0.y; a0.z += t0.z; a0.w += t0.w;
        }
        const float inv = 1.0f / fmaxf((float)(j1 - j0), 1.0f);
        float* p = &An[r * LDSTR + lane * 4];
        p[0] = (a0.x + a1.x) * inv;
        p[1] = (a0.y + a1.y) * inv;
        p[2] = (a0.z + a1.z) * inv;
        p[3] = (a0.w + a1.w) * inv;
    }
    __syncthreads();

    // WMMA phase
    const int hi = lane >> 4;   // selects K pair (A/B) and M half (C/D)
    const int lo = lane & 15;
    const int nTiles = DOUT / 16;

    for (int ct = wave; ct < nTiles; ct += 4) {
        const int col = ct * 16 + lo;
        const float* bSp = &WsT[(unsigned)col << 7];   // contiguous 128-float W column
        const float* bNp = &WnT[(unsigned)col << 7];
        v8f acc = {};
#pragma unroll
        for (int k = 0; k < 128; k += 4) {
            const int ka = k + hi * 2;
            v2f aS = *(const v2f*)&As[lo * LDSTR + ka];
            v2f aN = *(const v2f*)&An[lo * LDSTR + ka];
            v2f bS = *(const v2f*)&bSp[ka];  // single global_load_b64, imm offset
            v2f bN = *(const v2f*)&bNp[ka];
            acc = __builtin_amdgcn_wmma_f32_16x16x4_f32(false, aS, false, bS,
                                                        (short)0, acc, false, false);
            acc = __builtin_amdgcn_wmma_f32_16x16x4_f32(false, aN, false, bN,
                                                        (short)0, acc, false, false);
        }
        const float bv = bias[col];
#pragma unroll
        for (int r = 0; r < 8; ++r) {
            unsigned row = (unsigned)(rowBase + r + hi * 8);
            out[row * (unsigned)DOUT + (unsigned)col] = acc[r] + bv;
        }
    }
}

// ---------------- BN statistics / apply ----------------

__global__ __launch_bounds__(128)
void colstat128(const float* __restrict__ x, float* __restrict__ sums,
                float* __restrict__ sumsq, int n, int rowsPer) {
    int c  = threadIdx.x;
    int r0 = blockIdx.x * rowsPer;
    int r1 = min(r0 + rowsPer, n);
    float s = 0.f, q = 0.f;
    for (int r = r0; r < r1; ++r) {
        float v = x[(unsigned)r * 128u + (unsigned)c];
        s += v; q += v * v;
    }
    atomicAdd(&sums[c], s);
    atomicAdd(&sumsq[c], q);
}

__global__ __launch_bounds__(256)
void bn_relu128(float* __restrict__ x, const float* __restrict__ sums,
                const float* __restrict__ sumsq, const float* __restrict__ g,
                const float* __restrict__ be, int n) {
    unsigned i = blockIdx.x * 256u + threadIdx.x;
    if (i >= (unsigned)n * 128u) return;
    int c = (int)(i & 127u);
    float m = sums[c] / n;
    float v = sumsq[c] / n - m * m;
    float y = (x[i] - m) * rsqrtf(v + 1e-5f) * g[c] + be[c];
    x[i] = y > 0.f ? y : 0.f;
}

__global__ __launch_bounds__(64)
void colsum64(const float* __restrict__ x, float* __restrict__ sums, int n, int rowsPer) {
    int c  = threadIdx.x;
    int r0 = blockIdx.x * rowsPer;
    int r1 = min(r0 + rowsPer, n);
    float s = 0.f;
    for (int r = r0; r < r1; ++r) s += x[(unsigned)r * 64u + (unsigned)c];
    atomicAdd(&sums[c], s);
}

__global__ __launch_bounds__(64)
void logsoftmax64(const float* __restrict__ colsum, float* __restrict__ out, int n) {
    __shared__ float m[64];
    __shared__ float smax, slse;
    int j = threadIdx.x;
    m[j] = colsum[j] / (float)n;
    __syncthreads();
    if (j == 0) {
        float M = m[0];
        for (int k = 1; k < 64; ++k) M = fmaxf(M, m[k]);
        float s = 0.f;
        for (int k = 0; k < 64; ++k) s += expf(m[k] - M);
        smax = M;
        slse = logf(s);
    }
    __syncthreads();
    out[j] = m[j] - smax - slse;
}

// ---------------- host orchestration ----------------

extern "C" void kernel_launch(void* const* d_in, const int* in_sizes, int n_in,
                              void* d_out, int out_size, void* d_ws, size_t ws_size,
                              hipStream_t stream) {
    (void)n_in; (void)out_size; (void)ws_size;

    const float* x_in = (const float*)d_in[0];
    const int*   esrc = (const int*)d_in[1];
    const int*   edst = (const int*)d_in[2];
    const float* Ws0 = (const float*)d_in[3];
    const float* Wn0 = (const float*)d_in[4];
    const float* b0  = (const float*)d_in[5];
    const float* g0  = (const float*)d_in[6];
    const float* be0 = (const float*)d_in[7];
    const float* Ws1 = (const float*)d_in[8];
    const float* Wn1 = (const float*)d_in[9];
    const float* b1  = (const float*)d_in[10];
    const float* g1  = (const float*)d_in[11];
    const float* be1 = (const float*)d_in[12];
    const float* Ws2 = (const float*)d_in[13];
    const float* Wn2 = (const float*)d_in[14];
    const float* b2  = (const float*)d_in[15];

    const int N    = in_sizes[0] / 128;
    const int E    = in_sizes[1];
    const int OUTC = in_sizes[15];   // 64

    float* out_logp    = (float*)d_out;
    float* out_firstlx = out_logp + OUTC;
    float* out_feat    = out_firstlx + (size_t)N * 128;

    // workspace layout
    char*  wsb       = (char*)d_ws;
    int*   counts    = (int*)wsb;                    // N
    int*   cursor    = counts + N;                   // N
    int*   rowptr    = cursor + N;                   // N+1
    int*   srcSorted = rowptr + (N + 1);             // E
    float* h         = (float*)(srcSorted + E);      // N*128 (reused as N*64 for L2)
    float* sums      = h + (size_t)N * 128;          // 128
    float* sumsq     = sums + 128;                   // 128
    float* csum      = sumsq + 128;                  // 64
    float* WsT0      = csum + 64;                    // 128*128 each
    float* WnT0      = WsT0 + 128 * 128;
    float* WsT1      = WnT0 + 128 * 128;
    float* WnT1      = WsT1 + 128 * 128;
    float* WsT2      = WnT1 + 128 * 128;             // 64*128
    float* WnT2      = WsT2 + 64 * 128;

    const int TPB = 256;
    const long long NF = (long long)N * 128;
    const int rowTiles = N / 16;
    const int statRows = 256;
    const int statBlocks = (N + statRows - 1) / statRows;
    const int fBlocks = (int)((NF + TPB - 1) / TPB);
    const int eBlocks = (E + TPB - 1) / TPB;
    const int nBlocks = (N + TPB - 1) / TPB;

    // first_l_x = inputs
    {
        int n4 = (int)(NF / 4);
        copy_f4<<<(n4 + TPB - 1) / TPB, TPB, 0, stream>>>((const float4*)x_in,
                                                          (float4*)out_firstlx, n4);
    }

    // build CSR (dst-sorted edge list), once; reused by all three layers
    fill_i32<<<nBlocks, TPB, 0, stream>>>(counts, N, 0);
    hist_dst<<<eBlocks, TPB, 0, stream>>>(edst, counts, E);
    scan_counts<<<1, 1024, 0, stream>>>(counts, rowptr, N);
    fill_i32<<<nBlocks, TPB, 0, stream>>>(cursor, N, 0);
    build_csr<<<eBlocks, TPB, 0, stream>>>(esrc, edst, rowptr, cursor, srcSorted, E);

    // transpose weights for contiguous B-fragment loads
    {
        int wb128 = (128 * 128 + TPB - 1) / TPB;
        int wb64  = (128 * 64 + TPB - 1) / TPB;
        transpose_w<<<wb128, TPB, 0, stream>>>(Ws0, WsT0, 128);
        transpose_w<<<wb128, TPB, 0, stream>>>(Wn0, WnT0, 128);
        transpose_w<<<wb128, TPB, 0, stream>>>(Ws1, WsT1, 128);
        transpose_w<<<wb128, TPB, 0, stream>>>(Wn1, WnT1, 128);
        transpose_w<<<wb64,  TPB, 0, stream>>>(Ws2, WsT2, 64);
        transpose_w<<<wb64,  TPB, 0, stream>>>(Wn2, WnT2, 64);
    }

    // ---- layer 0: inputs -> h, BN+ReLU ----
    sage_fused<128><<<rowTiles, 128, 0, stream>>>(x_in, rowptr, srcSorted, WsT0, WnT0, b0, h);
    fill_f32<<<1, 256, 0, stream>>>(sums, 256, 0.f);
    colstat128<<<statBlocks, 128, 0, stream>>>(h, sums, sumsq, N, statRows);
    bn_relu128<<<fBlocks, TPB, 0, stream>>>(h, sums, sumsq, g0, be0, N);

    // ---- layer 1: h -> out_feat (feat output, layer-2 input), BN+ReLU ----
    sage_fused<128><<<rowTiles, 128, 0, stream>>>(h, rowptr, srcSorted, WsT1, WnT1, b1, out_feat);
    fill_f32<<<1, 256, 0, stream>>>(sums, 256, 0.f);
    colstat128<<<statBlocks, 128, 0, stream>>>(out_feat, sums, sumsq, N, statRows);
    bn_relu128<<<fBlocks, TPB, 0, stream>>>(out_feat, sums, sumsq, g1, be1, N);

    // ---- layer 2: out_feat -> h (N x 64) ----
    sage_fused<64><<<rowTiles, 128, 0, stream>>>(out_feat, rowptr, srcSorted, WsT2, WnT2, b2, h);

    // ---- column mean + log_softmax -> logp ----
    fill_f32<<<1, 64, 0, stream>>>(csum, 64, 0.f);
    colsum64<<<statBlocks, 64, 0, stream>>>(h, csum, N, statRows);
    logsoftmax64<<<1, 64, 0, stream>>>(csum, out_logp, N);
}